// GraphConvolution_38250978738649
// MI455X (gfx1250) — compile-verified
//
#include <hip/hip_runtime.h>

typedef float v2f __attribute__((ext_vector_type(2)));
typedef float v8f __attribute__((ext_vector_type(8)));

#define DIN  512
#define DOUT 512
#define KC   32                 // K-chunk staged per iteration
#define NCH  (DIN / KC)         // 16 chunks
#define BFLT (KC / 2 * 128 * 2) // floats per LDS B buffer (16KB)

// ---------------------------------------------------------------------------
// K1: support = X @ W  via V_WMMA_F32_16X16X4_F32.
// Block = 8 waves = 128 rows x 128 cols; all waves share one B slab.
// B staged K-pair-interleaved in LDS (b-frag = one ds_load_b64, no shuffles)
// via global_load_async_to_lds_b32, double-buffered so the async DMA of
// chunk i+1 overlaps the 64 WMMAs of chunk i (ASYNCcnt completes in order,
// so s_wait_asynccnt 16 == "previous chunk landed").
// ---------------------------------------------------------------------------
__global__ __launch_bounds__(256) void gcn_gemm_wmma(
    const float* __restrict__ X, const float* __restrict__ W,
    float* __restrict__ S, int Nrows)
{
    __shared__ float Bt[2][BFLT];          // 2 x 16 KB

    const int t    = threadIdx.x;
    const int wave = t >> 5;
    const int lane = t & 31;
    const int half = lane >> 4;            // 0: lanes 0-15, 1: lanes 16-31
    const int lm   = lane & 15;

    const int rb    = blockIdx.x >> 2;     // 128-row block
    const int c     = blockIdx.x & 3;      // 128-col block
    const int cbase = c * 128;

    const int row0 = rb * 128 + wave * 16; // this wave's 16-row tile
    int rowA = row0 + lm;                  // clamp so EXEC stays all-ones
    if (rowA > Nrows - 1) rowA = Nrows - 1;
    const float* Xrow = X + (size_t)rowA * DIN;

    // staging decode: shot s, thread t -> LDS float (s*256 + t) which is
    // pair p=s, col=t>>1, parity=t&1  (interleaved pair layout)
    const uint32_t ldsb0 = (uint32_t)(uintptr_t)(&Bt[0][0]) + (uint32_t)t * 4u;
    const uint32_t ldsb1 = (uint32_t)(uintptr_t)(&Bt[1][0]) + (uint32_t)t * 4u;
    const float*   gstag = W + (size_t)(t & 1) * DOUT + cbase + (t >> 1);

    v8f a0 = {}, a1 = {}, a2 = {}, a3 = {}, a4 = {}, a5 = {}, a6 = {}, a7 = {};

    // issue one chunk's 16 async b32 copies (rows kbase..kbase+31)
    auto stage = [&](int kbase, int buf) {
        const float*   g  = gstag + (size_t)kbase * DOUT;
        const uint32_t lb = buf ? ldsb1 : ldsb0;
#pragma unroll
        for (int s = 0; s < 16; ++s) {
            asm volatile("global_load_async_to_lds_b32 %0, %1, off"
                         :: "v"(lb + (uint32_t)s * 1024u),
                            "v"(g + (size_t)(2 * s) * DOUT)
                         : "memory");
        }
    };

    int buf = 0;
    stage(0, 0);

#pragma unroll 1
    for (int i = 0; i < NCH; ++i) {
        const int k = i * KC;
        if (i + 1 < NCH) {
            stage(k + KC, buf ^ 1);
            asm volatile("s_wait_asynccnt 0x10" ::: "memory"); // chunk i landed
        } else {
            asm volatile("s_wait_asynccnt 0x0" ::: "memory");
        }
        __syncthreads();

        const float* Bc = &Bt[buf][0];
#pragma unroll
        for (int kk = 0; kk < KC; kk += 4) {
            // A-frag 16x4: lanes 0-15 K=kk..kk+1, lanes 16-31 K=kk+2..kk+3
            v2f a = *(const v2f*)(Xrow + k + kk + 2 * half);

            // B-frags: pair p holds rows (2p, 2p+1) interleaved per column
            const int    p  = (kk >> 1) + half;
            const float* Bp = Bc + ((size_t)p * 128 + lm) * 2;
            v2f b0 = *(const v2f*)(Bp +   0);
            v2f b1 = *(const v2f*)(Bp +  32);
            v2f b2 = *(const v2f*)(Bp +  64);
            v2f b3 = *(const v2f*)(Bp +  96);
            v2f b4 = *(const v2f*)(Bp + 128);
            v2f b5 = *(const v2f*)(Bp + 160);
            v2f b6 = *(const v2f*)(Bp + 192);
            v2f b7 = *(const v2f*)(Bp + 224);

            a0 = __builtin_amdgcn_wmma_f32_16x16x4_f32(false, a, false, b0, (short)0, a0, false, false);
            a1 = __builtin_amdgcn_wmma_f32_16x16x4_f32(false, a, false, b1, (short)0, a1, false, false);
            a2 = __builtin_amdgcn_wmma_f32_16x16x4_f32(false, a, false, b2, (short)0, a2, false, false);
            a3 = __builtin_amdgcn_wmma_f32_16x16x4_f32(false, a, false, b3, (short)0, a3, false, false);
            a4 = __builtin_amdgcn_wmma_f32_16x16x4_f32(false, a, false, b4, (short)0, a4, false, false);
            a5 = __builtin_amdgcn_wmma_f32_16x16x4_f32(false, a, false, b5, (short)0, a5, false, false);
            a6 = __builtin_amdgcn_wmma_f32_16x16x4_f32(false, a, false, b6, (short)0, a6, false, false);
            a7 = __builtin_amdgcn_wmma_f32_16x16x4_f32(false, a, false, b7, (short)0, a7, false, false);
        }
        __syncthreads();
        buf ^= 1;
    }

    // C/D layout: VGPR v -> row (row0 + half*8 + v), lane -> col (cbase + tl*16 + lm)
    const int rowS = row0 + half * 8;
    float* Sr = S + (size_t)rowS * DOUT + cbase + lm;
#pragma unroll
    for (int v = 0; v < 8; ++v) {
        if (rowS + v < Nrows) {
            float* Sv = Sr + (size_t)v * DOUT;
            Sv[  0] = a0[v];
            Sv[ 16] = a1[v];
            Sv[ 32] = a2[v];
            Sv[ 48] = a3[v];
            Sv[ 64] = a4[v];
            Sv[ 80] = a5[v];
            Sv[ 96] = a6[v];
            Sv[112] = a7[v];
        }
    }
}

// ---------------------------------------------------------------------------
// K0: zero int buffer
// ---------------------------------------------------------------------------
__global__ void gcn_zero_i32(int* __restrict__ p, int n)
{
    int i = blockIdx.x * blockDim.x + threadIdx.x;
    if (i < n) p[i] = 0;
}

// ---------------------------------------------------------------------------
// K2: histogram of edge destinations
// ---------------------------------------------------------------------------
__global__ void gcn_count_dst(const int* __restrict__ dst, int E,
                              int* __restrict__ counts)
{
    int e = blockIdx.x * blockDim.x + threadIdx.x;
    if (e < E) atomicAdd(&counts[dst[e]], 1);
}

// ---------------------------------------------------------------------------
// K3: single-block exclusive scan of counts -> offsets (+ cursor copy)
// ---------------------------------------------------------------------------
__global__ __launch_bounds__(1024) void gcn_scan(
    const int* __restrict__ counts, int* __restrict__ offsets,
    int* __restrict__ cursor, int N)
{
    __shared__ int part[1024];
    const int t     = threadIdx.x;
    const int chunk = (N + 1023) >> 10;
    int lo = t * chunk;
    int hi = lo + chunk; if (hi > N) hi = N;

    int s = 0;
    for (int i = lo; i < hi; ++i) s += counts[i];
    part[t] = s;
    __syncthreads();
    for (int off = 1; off < 1024; off <<= 1) {
        int v = (t >= off) ? part[t - off] : 0;
        __syncthreads();
        part[t] += v;
        __syncthreads();
    }
    int run = part[t] - s;  // exclusive prefix of this chunk
    for (int i = lo; i < hi; ++i) {
        offsets[i] = run;
        cursor[i]  = run;
        run += counts[i];
    }
    if (t == 1023) offsets[N] = part[1023];
}

// ---------------------------------------------------------------------------
// K4: permute edges into CSR order; pack (src, val) -> int2 so the
// aggregation kernel can fetch edges with scalar loads.
// ---------------------------------------------------------------------------
__global__ void gcn_scatter_edges(const int* __restrict__ dst,
                                  const int* __restrict__ src,
                                  const float* __restrict__ val, int E,
                                  int* __restrict__ cursor,
                                  int2* __restrict__ psv)
{
    int e = blockIdx.x * blockDim.x + threadIdx.x;
    if (e < E) {
        int d   = dst[e];
        int pos = atomicAdd(&cursor[d], 1);
        int2 p;
        p.x = src[e];
        p.y = __float_as_int(val[e]);
        psv[pos] = p;
    }
}

// ---------------------------------------------------------------------------
// K5: per-destination-row aggregation. One wave per node; each lane owns
// 4 float4 column slices (coalesced 512B vector loads, L2-resident since
// support = 102MB < 192MB L2). Edge (src,val) pairs fetched via uniform
// (readfirstlane) indices -> scalar loads; loop unrolled x2 so 8
// independent 512B gathers are in flight per wave.
// ---------------------------------------------------------------------------
__global__ __launch_bounds__(256) void gcn_spmm_rows(
    const float4* __restrict__ S4, const int* __restrict__ offsets,
    const int2* __restrict__ psv, const float* __restrict__ bias,
    float4* __restrict__ out4, int N)
{
    const int wave = threadIdx.x >> 5;
    const int lane = threadIdx.x & 31;
    const int d    = blockIdx.x * 8 + wave;
    if (d >= N) return;

    const int beg = __builtin_amdgcn_readfirstlane(offsets[d]);
    const int end = __builtin_amdgcn_readfirstlane(offsets[d + 1]);

    float4 a0 = {0.f, 0.f, 0.f, 0.f}, a1 = a0, a2 = a0, a3 = a0;

    int e = beg;
    for (; e + 2 <= end; e += 2) {
        const int2 p0 = psv[e];
        const int2 p1 = psv[e + 1];
        const float w0 = __int_as_float(p0.y);
        const float w1 = __int_as_float(p1.y);
        const float4* R0 = S4 + (size_t)p0.x * 128;
        const float4* R1 = S4 + (size_t)p1.x * 128;
        float4 x0 = R0[lane +  0], x1 = R0[lane + 32];
        float4 x2 = R0[lane + 64], x3 = R0[lane + 96];
        float4 y0 = R1[lane +  0], y1 = R1[lane + 32];
        float4 y2 = R1[lane + 64], y3 = R1[lane + 96];
        a0.x = fmaf(w0, x0.x, a0.x); a0.y = fmaf(w0, x0.y, a0.y);
        a0.z = fmaf(w0, x0.z, a0.z); a0.w = fmaf(w0, x0.w, a0.w);
        a1.x = fmaf(w0, x1.x, a1.x); a1.y = fmaf(w0, x1.y, a1.y);
        a1.z = fmaf(w0, x1.z, a1.z); a1.w = fmaf(w0, x1.w, a1.w);
        a2.x = fmaf(w0, x2.x, a2.x); a2.y = fmaf(w0, x2.y, a2.y);
        a2.z = fmaf(w0, x2.z, a2.z); a2.w = fmaf(w0, x2.w, a2.w);
        a3.x = fmaf(w0, x3.x, a3.x); a3.y = fmaf(w0, x3.y, a3.y);
        a3.z = fmaf(w0, x3.z, a3.z); a3.w = fmaf(w0, x3.w, a3.w);
        a0.x = fmaf(w1, y0.x, a0.x); a0.y = fmaf(w1, y0.y, a0.y);
        a0.z = fmaf(w1, y0.z, a0.z); a0.w = fmaf(w1, y0.w, a0.w);
        a1.x = fmaf(w1, y1.x, a1.x); a1.y = fmaf(w1, y1.y, a1.y);
        a1.z = fmaf(w1, y1.z, a1.z); a1.w = fmaf(w1, y1.w, a1.w);
        a2.x = fmaf(w1, y2.x, a2.x); a2.y = fmaf(w1, y2.y, a2.y);
        a2.z = fmaf(w1, y2.z, a2.z); a2.w = fmaf(w1, y2.w, a2.w);
        a3.x = fmaf(w1, y3.x, a3.x); a3.y = fmaf(w1, y3.y, a3.y);
        a3.z = fmaf(w1, y3.z, a3.z); a3.w = fmaf(w1, y3.w, a3.w);
    }
    if (e < end) {
        const int2 p0 = psv[e];
        const float w0 = __int_as_float(p0.y);
        const float4* R0 = S4 + (size_t)p0.x * 128;
        float4 x0 = R0[lane +  0], x1 = R0[lane + 32];
        float4 x2 = R0[lane + 64], x3 = R0[lane + 96];
        a0.x = fmaf(w0, x0.x, a0.x); a0.y = fmaf(w0, x0.y, a0.y);
        a0.z = fmaf(w0, x0.z, a0.z); a0.w = fmaf(w0, x0.w, a0.w);
        a1.x = fmaf(w0, x1.x, a1.x); a1.y = fmaf(w0, x1.y, a1.y);
        a1.z = fmaf(w0, x1.z, a1.z); a1.w = fmaf(w0, x1.w, a1.w);
        a2.x = fmaf(w0, x2.x, a2.x); a2.y = fmaf(w0, x2.y, a2.y);
        a2.z = fmaf(w0, x2.z, a2.z); a2.w = fmaf(w0, x2.w, a2.w);
        a3.x = fmaf(w0, x3.x, a3.x); a3.y = fmaf(w0, x3.y, a3.y);
        a3.z = fmaf(w0, x3.z, a3.z); a3.w = fmaf(w0, x3.w, a3.w);
    }

    const float4* b4 = (const float4*)bias;
    float4 r0 = b4[lane +  0], r1 = b4[lane + 32];
    float4 r2 = b4[lane + 64], r3 = b4[lane + 96];
    r0.x += a0.x; r0.y += a0.y; r0.z += a0.z; r0.w += a0.w;
    r1.x += a1.x; r1.y += a1.y; r1.z += a1.z; r1.w += a1.w;
    r2.x += a2.x; r2.y += a2.y; r2.z += a2.z; r2.w += a2.w;
    r3.x += a3.x; r3.y += a3.y; r3.z += a3.z; r3.w += a3.w;

    float4* Or = out4 + (size_t)d * 128;
    Or[lane +  0] = r0;
    Or[lane + 32] = r1;
    Or[lane + 64] = r2;
    Or[lane + 96] = r3;
}

// ---------------------------------------------------------------------------
// Host launcher
// ---------------------------------------------------------------------------
extern "C" void kernel_launch(void* const* d_in, const int* in_sizes, int n_in,
                              void* d_out, int out_size, void* d_ws, size_t ws_size,
                              hipStream_t stream)
{
    (void)n_in; (void)out_size; (void)ws_size;

    const float* X    = (const float*)d_in[0];
    const float* W    = (const float*)d_in[1];
    const float* bias = (const float*)d_in[2];
    const int*   ei   = (const int*)d_in[3];   // [2, E]: row0 = dst, row1 = src
    const float* ev   = (const float*)d_in[4];

    const int N = in_sizes[0] / DIN;
    const int E = in_sizes[3] / 2;
    const int* dst = ei;
    const int* src = ei + E;

    // workspace carve-up (all slices 16B-aligned)
    char*  ws      = (char*)d_ws;
    float* S       = (float*)ws;  ws += (size_t)N * DOUT * sizeof(float);
    int*   counts  = (int*)ws;    ws += (size_t)(N + 4) * sizeof(int);
    int*   offsets = (int*)ws;    ws += (size_t)(N + 4) * sizeof(int);
    int*   cursor  = (int*)ws;    ws += (size_t)(N + 4) * sizeof(int);
    int2*  psv     = (int2*)ws;

    // K0: clear histogram
    gcn_zero_i32<<<(N + 256) / 256, 256, 0, stream>>>(counts, N + 1);

    // K1: support = X @ W (WMMA fp32, double-buffered async-LDS B staging).
    // Block tile = 128 rows x 128 cols -> grid = ceil(N/128) * 4 col blocks.
    gcn_gemm_wmma<<<((N + 127) / 128) * 4, 256, 0, stream>>>(X, W, S, N);

    // K2..K4: build CSR-by-destination
    gcn_count_dst<<<(E + 255) / 256, 256, 0, stream>>>(dst, E, counts);
    gcn_scan<<<1, 1024, 0, stream>>>(counts, offsets, cursor, N);
    gcn_scatter_edges<<<(E + 255) / 256, 256, 0, stream>>>(dst, src, ev, E,
                                                           cursor, psv);

    // K5: out[d] = sum_e val*support[src] + bias  (one wave per node)
    gcn_spmm_rows<<<(N + 7) / 8, 256, 0, stream>>>((const float4*)S, offsets,
                                                   psv, bias,
                                                   (float4*)d_out, N);
}